// _TrainLevel_23338852286541
// MI455X (gfx1250) — compile-verified
//
#include <hip/hip_runtime.h>
#include <stdint.h>

#define HASH_MASK ((1u << 22) - 1u)   // BUCKETS = 2^22
#define BLK 256

typedef float v4f __attribute__((ext_vector_type(4)));

// ---- gfx1250 async global->LDS DMA (ASYNCcnt-tracked), per-lane addresses ----
__device__ __forceinline__ void async_copy_b128_g2l(unsigned lds_byte_addr,
                                                    const void* gsrc) {
  asm volatile("global_load_async_to_lds_b128 %0, %1, off"
               :: "v"(lds_byte_addr),
                  "v"((unsigned long long)(uintptr_t)gsrc)
               : "memory");
}

__device__ __forceinline__ void wait_asynccnt0() {
#if __has_builtin(__builtin_amdgcn_s_wait_asynccnt)
  __builtin_amdgcn_s_wait_asynccnt(0);
#else
  asm volatile("s_wait_asynccnt 0" ::: "memory");
#endif
}

__global__ __launch_bounds__(BLK) void hashgrid_trilerp_kernel(
    const float* __restrict__ pts,          // (N,3) f32
    const float* __restrict__ feats,        // (2^22, 8) f32 — 128 MB, L2-resident
    const unsigned* __restrict__ primes,    // (3,) u32
    float* __restrict__ out,                // (N,8) f32
    int n)
{
  __shared__ __align__(16) float s_pts[BLK * 3];   // 3 KB point tile

  const int tid = threadIdx.x;
  const long long blockBase = (long long)blockIdx.x * BLK;
  const long long byteBase  = blockBase * 12ll;
  const long long totalB    = (long long)n * 12ll;
  long long remll = totalB - byteBase;
  const int bytes = remll > (long long)(BLK * 12) ? (BLK * 12) : (int)remll;

  // Stage the block's AoS float3 tile into LDS with wide async DMA transfers.
  {
    const int chunkOff = tid * 16;
    if (chunkOff + 16 <= bytes) {
      async_copy_b128_g2l((unsigned)(uintptr_t)s_pts + (unsigned)chunkOff,
                          (const char*)pts + byteBase + chunkOff);
    } else if (chunkOff < bytes) {
      // ragged tail (< 16 B): plain dword copies
      for (int b = chunkOff; b < bytes; b += 4)
        *(float*)((char*)s_pts + b) =
            *(const float*)((const char*)pts + byteBase + b);
    }
  }
  wait_asynccnt0();      // per-wave: async writes to LDS are visible
  __syncthreads();       // cross-wave visibility of the tile

  const long long i = blockBase + tid;
  if (i >= n) return;

  // --- grid coords: pts/RES == pts*1024.0f exactly (RES = 2^-10) ---
  const float qx = s_pts[3 * tid + 0] * 1024.0f;
  const float qy = s_pts[3 * tid + 1] * 1024.0f;
  const float qz = s_pts[3 * tid + 2] * 1024.0f;
  const float bx = floorf(qx), by = floorf(qy), bz = floorf(qz);
  const float fx = qx - bx, fy = qy - by, fz = qz - bz;

  // --- spatial hash (uint32 wraparound == reference; 2^22 | 2^32) ---
  const unsigned P0 = primes[0], P1 = primes[1], P2 = primes[2];
  const unsigned ux = (unsigned)(int)bx;
  const unsigned uy = (unsigned)(int)by;
  const unsigned uz = (unsigned)(int)bz;
  const unsigned h = ux * P0 + uy * P1 + uz * P2;

  unsigned vid[8];
  vid[0] = h;            vid[1] = h + P0;
  vid[2] = h + P1;       vid[3] = h + P0 + P1;
  vid[4] = h + P2;       vid[5] = h + P0 + P2;
  vid[6] = h + P1 + P2;  vid[7] = h + P0 + P1 + P2;
#pragma unroll
  for (int c = 0; c < 8; ++c) vid[c] &= HASH_MASK;

  // --- trilinear weights, CORNER_OFFSETS order ---
  const float gx0 = 1.0f - fx, gx1 = fx;
  const float gy0 = 1.0f - fy, gy1 = fy;
  const float gz0 = 1.0f - fz, gz1 = fz;
  float w[8];
  w[0] = gx0 * gy0 * gz0;  w[1] = gx1 * gy0 * gz0;
  w[2] = gx0 * gy1 * gz0;  w[3] = gx1 * gy1 * gz0;
  w[4] = gx0 * gy0 * gz1;  w[5] = gx1 * gy0 * gz1;
  w[6] = gx0 * gy1 * gz1;  w[7] = gx1 * gy1 * gz1;

  // --- 16 independent b128 gathers (L2-resident table), fused accumulate ---
  v4f acc0 = (v4f)0.0f, acc1 = (v4f)0.0f;
#pragma unroll
  for (int c = 0; c < 8; ++c) {
    const v4f* fp = (const v4f*)(feats + ((size_t)vid[c] << 3));
    const v4f a = fp[0];
    const v4f b = fp[1];
    const float wc = w[c];
    acc0 += wc * a;
    acc1 += wc * b;
  }

  // --- streaming output: non-temporal b128 stores (don't evict the table) ---
  v4f* op = (v4f*)(out + ((size_t)i << 3));
  __builtin_nontemporal_store(acc0, op);
  __builtin_nontemporal_store(acc1, op + 1);
}

extern "C" void kernel_launch(void* const* d_in, const int* in_sizes, int n_in,
                              void* d_out, int out_size, void* d_ws, size_t ws_size,
                              hipStream_t stream) {
  const float*    pts    = (const float*)d_in[0];
  const float*    feats  = (const float*)d_in[1];
  const unsigned* primes = (const unsigned*)d_in[2];
  float*          out    = (float*)d_out;

  const int n = in_sizes[0] / 3;
  const int blocks = (n + BLK - 1) / BLK;
  hashgrid_trilerp_kernel<<<blocks, BLK, 0, stream>>>(pts, feats, primes, out, n);
}